// StandardGCN_7103875908245
// MI455X (gfx1250) — compile-verified
//
#include <hip/hip_runtime.h>
#include <hip/hip_bf16.h>

// ---------------------------------------------------------------------------
// GCN: h = relu( D^-1/2 (A+I) D^-1/2 (x @ W) + b );  z = segpool(h);  out = z@Wc + bc
// Sizes: N=100000, E=3.2M, D_IN=384, D_H=64, G=512
// ---------------------------------------------------------------------------

typedef float v2f __attribute__((ext_vector_type(2)));
typedef float v8f __attribute__((ext_vector_type(8)));

#define D_IN 384
#define D_H  64
#define NGRAPHS 512

// ---------------- zero workspace region ----------------
__global__ void zero_f32(float* __restrict__ p, size_t n) {
    size_t i = (size_t)blockIdx.x * blockDim.x + threadIdx.x;
    if (i < n) p[i] = 0.0f;
}

// ---------------- degree over dst (edges only; +1 self loop added later) ----
__global__ void deg_kernel(const int* __restrict__ dst, float* __restrict__ deg, int E) {
    int e = blockIdx.x * blockDim.x + threadIdx.x;
    if (e < E) atomicAdd(&deg[dst[e]], 1.0f);
}

__global__ void dinv_kernel(const float* __restrict__ deg, float* __restrict__ dinv, int N) {
    int i = blockIdx.x * blockDim.x + threadIdx.x;
    if (i < N) dinv[i] = rsqrtf(deg[i] + 1.0f);   // deg>=1 always (self loop)
}

// ---------------- GEMM: H[N,64] = X[N,384] @ W[384,64] via V_WMMA_F32_16X16X4_F32
// block = 256 threads = 8 waves; wave w owns rows [blockIdx*128 + 16w, +16), all 64 cols.
__global__ __launch_bounds__(256)
void gemm_wmma_f32(const float* __restrict__ X, const float* __restrict__ W,
                   float* __restrict__ H, int N) {
    const int wave = threadIdx.x >> 5;
    const int lane = threadIdx.x & 31;
    const int row0 = blockIdx.x * 128 + wave * 16;
    if (row0 >= N) return;                       // wave-uniform exit, EXEC stays full

    const int m  = lane & 15;                    // A: row-in-tile; B/C: col-in-tile
    const int kh = lane >> 4;                    // 0: K pair {0,1}; 1: K pair {2,3}

    int row = row0 + m;
    if (row >= N) row = N - 1;                   // clamp: corrupt rows never stored
    const float* __restrict__ xrow = X + (size_t)row * D_IN;

    v8f acc0 = {}, acc1 = {}, acc2 = {}, acc3 = {};

    for (int k0 = 0; k0 < D_IN; k0 += 4) {
        // A fragment (16x4 f32): lanes 0-15 -> K=k0,k0+1 ; lanes 16-31 -> K=k0+2,k0+3
        v2f a;
        a.x = xrow[k0 + 2 * kh + 0];
        a.y = xrow[k0 + 2 * kh + 1];
        // B fragments (4x16 f32), W row-major [K=384, N=64]; lane column = m
        const float* __restrict__ wb = W + (size_t)(k0 + 2 * kh) * D_H + m;
        v2f b0, b1, b2, b3;
        b0.x = wb[0];       b0.y = wb[D_H + 0];
        b1.x = wb[16];      b1.y = wb[D_H + 16];
        b2.x = wb[32];      b2.y = wb[D_H + 32];
        b3.x = wb[48];      b3.y = wb[D_H + 48];

        acc0 = __builtin_amdgcn_wmma_f32_16x16x4_f32(false, a, false, b0, (short)0, acc0, false, false);
        acc1 = __builtin_amdgcn_wmma_f32_16x16x4_f32(false, a, false, b1, (short)0, acc1, false, false);
        acc2 = __builtin_amdgcn_wmma_f32_16x16x4_f32(false, a, false, b2, (short)0, acc2, false, false);
        acc3 = __builtin_amdgcn_wmma_f32_16x16x4_f32(false, a, false, b3, (short)0, acc3, false, false);
    }

    // C/D layout: lane n = m (per 16-lane half); VGPR r -> M = r + 8*kh
    #pragma unroll
    for (int r = 0; r < 8; ++r) {
        const int orow = row0 + r + 8 * kh;
        if (orow < N) {
            float* __restrict__ o = H + (size_t)orow * D_H;
            o[ 0 + m] = acc0[r];
            o[16 + m] = acc1[r];
            o[32 + m] = acc2[r];
            o[48 + m] = acc3[r];
        }
    }
}

// ---------------- scatter: agg[dst] += h[src] * dinv[src]*dinv[dst]
// 16 lanes per edge, float4 per lane (64 feats). agg & h are L2-resident (25.6 MB each).
__global__ __launch_bounds__(256)
void scatter_kernel(const int* __restrict__ src, const int* __restrict__ dst,
                    const float* __restrict__ dinv, const float* __restrict__ H,
                    float* __restrict__ agg, int E) {
    long long t = (long long)blockIdx.x * blockDim.x + threadIdx.x;
    int e  = (int)(t >> 4);
    int d4 = ((int)t & 15) * 4;
    if (e >= E) return;
    const int s = src[e];
    const int d = dst[e];
    const float nrm = dinv[s] * dinv[d];
    const float4 hv = *(const float4*)(H + (size_t)s * D_H + d4);
    float* __restrict__ o = agg + (size_t)d * D_H + d4;
    atomicAdd(o + 0, hv.x * nrm);
    atomicAdd(o + 1, hv.y * nrm);
    atomicAdd(o + 2, hv.z * nrm);
    atomicAdd(o + 3, hv.w * nrm);
}

// ---------------- relu(agg + selfloop + bias) then pool into z[graph, feat]
__global__ __launch_bounds__(256)
void relu_pool_kernel(const float* __restrict__ agg, const float* __restrict__ H,
                      const float* __restrict__ dinv, const float* __restrict__ bgcn,
                      const int* __restrict__ batch, float* __restrict__ z, int N) {
    long long t = (long long)blockIdx.x * blockDim.x + threadIdx.x;
    int i = (int)(t >> 6);
    int d = (int)t & 63;
    if (i >= N) return;
    const float di = dinv[i];
    float v = agg[(size_t)i * D_H + d] + H[(size_t)i * D_H + d] * di * di + bgcn[d];
    v = v > 0.0f ? v : 0.0f;
    atomicAdd(&z[(size_t)batch[i] * D_H + d], v);
}

// ---------------- classifier: out[g] = z[g,:] . Wc + bc
__global__ void cls_kernel(const float* __restrict__ z, const float* __restrict__ Wc,
                           const float* __restrict__ bc, float* __restrict__ out) {
    int g = blockIdx.x * blockDim.x + threadIdx.x;
    if (g >= NGRAPHS) return;
    float s = bc[0];
    #pragma unroll 8
    for (int d = 0; d < D_H; ++d) s += z[(size_t)g * D_H + d] * Wc[d];
    out[g] = s;
}

extern "C" void kernel_launch(void* const* d_in, const int* in_sizes, int n_in,
                              void* d_out, int out_size, void* d_ws, size_t ws_size,
                              hipStream_t stream) {
    const float* x_sem   = (const float*)d_in[0];
    const int*   eidx    = (const int*)d_in[1];
    const int*   batch   = (const int*)d_in[2];
    const float* W_gcn   = (const float*)d_in[3];
    const float* b_gcn   = (const float*)d_in[4];
    const float* W_cls   = (const float*)d_in[5];
    const float* b_cls   = (const float*)d_in[6];
    float*       out     = (float*)d_out;

    const int N = in_sizes[2];            // batch_index length
    const int E = in_sizes[1] / 2;        // edge_index is [2, E]

    const int* e_src = eidx;              // edge_index[0]
    const int* e_dst = eidx + E;          // edge_index[1]

    // workspace layout (floats): [ h : N*64 | agg : N*64 | deg : N | z : 512*64 | dinv : N ]
    float* ws   = (float*)d_ws;
    float* h    = ws;
    float* agg  = h   + (size_t)N * D_H;
    float* deg  = agg + (size_t)N * D_H;
    float* z    = deg + (size_t)N;
    float* dinv = z   + (size_t)NGRAPHS * D_H;

    // 1) zero [agg | deg | z] contiguously (h, dinv fully overwritten)
    {
        size_t nz = (size_t)N * D_H + (size_t)N + (size_t)NGRAPHS * D_H;
        zero_f32<<<(unsigned)((nz + 255) / 256), 256, 0, stream>>>(agg, nz);
    }
    // 2) degree over edge dst
    deg_kernel<<<(E + 255) / 256, 256, 0, stream>>>(e_dst, deg, E);
    // 3) dinv = rsqrt(deg + 1)   (self loop)
    dinv_kernel<<<(N + 255) / 256, 256, 0, stream>>>(deg, dinv, N);
    // 4) h = x @ W  (WMMA f32)
    gemm_wmma_f32<<<(N + 127) / 128, 256, 0, stream>>>(x_sem, W_gcn, h, N);
    // 5) edge scatter with symmetric norm
    {
        long long threads = (long long)E * 16;
        scatter_kernel<<<(unsigned)((threads + 255) / 256), 256, 0, stream>>>(
            e_src, e_dst, dinv, h, agg, E);
    }
    // 6) self-loop + bias + relu + pool
    {
        long long threads = (long long)N * D_H;
        relu_pool_kernel<<<(unsigned)((threads + 255) / 256), 256, 0, stream>>>(
            agg, h, dinv, b_gcn, batch, z, N);
    }
    // 7) classifier
    cls_kernel<<<2, 256, 0, stream>>>(z, W_cls, b_cls, out);
}